// Quantization_43319040147736
// MI455X (gfx1250) — compile-verified
//
#include <hip/hip_runtime.h>

typedef __attribute__((ext_vector_type(2))) float v2f;
typedef __attribute__((ext_vector_type(8))) float v8f;

namespace {
constexpr int kB = 8192;
constexpr int kD = 768;
constexpr int kM = 96;
constexpr int kK = 256;   // codes per subvector
constexpr int kDsub = 8;  // subvector length
constexpr int kCbStride = 9;       // padded LDS row stride (floats) -> conflict-free
constexpr int kWavesPerBlock = 8;  // 256 threads
constexpr int kRowsPerWave = 16;
}

__global__ __launch_bounds__(256)
void pq_assign_wmma(const float* __restrict__ vecs,
                    const float* __restrict__ codebook,
                    float* __restrict__ out) {
  __shared__ float s_cb[kK * kCbStride];  // codebook[m], padded rows (9.2 KB)
  __shared__ float s_c2[kK];              // ||c||^2 per code

  const int m    = blockIdx.y;
  const int tid  = threadIdx.x;
  const int wave = tid >> 5;
  const int lane = tid & 31;
  const int half = lane >> 4;   // 0: lanes 0-15, 1: lanes 16-31
  const int l16  = lane & 15;

  // ---- stage codebook[m] (256 x 8 fp32 = 8 KB) into LDS ----
  const float* cbm = codebook + (size_t)m * (kK * kDsub);
  #pragma unroll
  for (int i = 0; i < (kK * kDsub) / 256; ++i) {
    int idx = tid + i * 256;
    s_cb[(idx >> 3) * kCbStride + (idx & 7)] = cbm[idx];
  }
  __syncthreads();

  // ---- per-code squared norms (one code per thread) ----
  {
    float acc = 0.f;
    #pragma unroll
    for (int d = 0; d < kDsub; ++d) {
      float c = s_cb[tid * kCbStride + d];
      acc += c * c;
    }
    s_c2[tid] = acc;
  }
  __syncthreads();

  // ---- A fragments: 16 b-rows of subvector m; K=8 as two K=4 WMMA steps ----
  // ISA 16x4 fp32 A layout: VGPR0 = K0 (lanes 0-15) | K2 (lanes 16-31); VGPR1 = K1 | K3
  const int b_base = (blockIdx.x * kWavesPerBlock + wave) * kRowsPerWave;
  const float* vrow = vecs + (size_t)(b_base + l16) * kD + m * kDsub;
  v2f a0 = *(const v2f*)(vrow + half * 2);      // K = {0,1} | {2,3}
  v2f a1 = *(const v2f*)(vrow + 4 + half * 2);  // K = {4,5} | {6,7}

  float best[8];
  int   bidx[8];
  #pragma unroll
  for (int r = 0; r < 8; ++r) { best[r] = -__builtin_huge_valf(); bidx[r] = 0; }

  #pragma unroll
  for (int nt = 0; nt < kK / 16; ++nt) {
    const int kcol = nt * 16 + l16;  // code (column N) handled by this lane
    const float* cbrow = &s_cb[kcol * kCbStride];
    v2f b0, b1;  // 4x16 fp32 B fragment, mirrored layout of A
    b0.x = cbrow[half * 2 + 0];     b0.y = cbrow[half * 2 + 1];
    b1.x = cbrow[4 + half * 2 + 0]; b1.y = cbrow[4 + half * 2 + 1];

    v8f c = {};
    c = __builtin_amdgcn_wmma_f32_16x16x4_f32(false, a0, false, b0,
                                              (short)0, c, false, false);
    c = __builtin_amdgcn_wmma_f32_16x16x4_f32(false, a1, false, b1,
                                              (short)0, c, false, false);

    const float c2k = s_c2[kcol];
    #pragma unroll
    for (int r = 0; r < 8; ++r) {
      float val = 2.0f * c[r] - c2k;  // == proba + ||v||^2 (row-constant dropped)
      if (val > best[r]) { best[r] = val; bidx[r] = kcol; }
    }
  }

  // ---- argmax across the 16 lanes of each half-wave (xor masks keep halves apart) ----
  #pragma unroll
  for (int off = 8; off >= 1; off >>= 1) {
    #pragma unroll
    for (int r = 0; r < 8; ++r) {
      float ov = __shfl_xor(best[r], off, 32);
      int   oi = __shfl_xor(bidx[r], off, 32);
      if (ov > best[r] || (ov == best[r] && oi < bidx[r])) {
        best[r] = ov; bidx[r] = oi;
      }
    }
  }

  // ---- gather winning codewords: lane 0 -> rows 0..7, lane 16 -> rows 8..15 ----
  if (l16 == 0) {
    #pragma unroll
    for (int r = 0; r < 8; ++r) {
      const int row = b_base + half * 8 + r;  // C/D layout: VGPR r = row r / r+8
      float* o = out + (size_t)row * kD + m * kDsub;
      const float* src = &s_cb[bidx[r] * kCbStride];
      #pragma unroll
      for (int d = 0; d < kDsub; ++d) o[d] = src[d];
    }
  }
}

extern "C" void kernel_launch(void* const* d_in, const int* in_sizes, int n_in,
                              void* d_out, int out_size, void* d_ws, size_t ws_size,
                              hipStream_t stream) {
  const float* vecs     = (const float*)d_in[0];
  const float* codebook = (const float*)d_in[1];
  float* out = (float*)d_out;
  dim3 grid(kB / (kWavesPerBlock * kRowsPerWave), kM);  // (64, 96)
  pq_assign_wmma<<<grid, 256, 0, stream>>>(vecs, codebook, out);
}